// GCN_19232863552093
// MI455X (gfx1250) — compile-verified
//
#include <hip/hip_runtime.h>
#include <math.h>

#define NN   50000
#define HIDF 128
#define EE   800000
#define ET   (EE + NN)        // 850000 edges incl. self loops
#define SLOPE 0.2f
#define MTILES 5              // 5 x 16 = 80 rows per block; 50000 = 625*80 exactly

typedef __attribute__((ext_vector_type(2))) float v2f;
typedef __attribute__((ext_vector_type(8))) float v8f;

// monotonic float -> uint encoding so atomicMax(uint) == float max
__device__ __forceinline__ unsigned fenc(float f) {
  unsigned u = __float_as_uint(f);
  return (u & 0x80000000u) ? ~u : (u | 0x80000000u);
}
__device__ __forceinline__ float fdec(unsigned u) {
  unsigned b = (u & 0x80000000u) ? (u & 0x7FFFFFFFu) : ~u;
  return __uint_as_float(b);
}
__device__ __forceinline__ float lrelu(float x) { return x > 0.f ? x : SLOPE * x; }
__device__ __forceinline__ float wred(float v) {
#pragma unroll
  for (int o = 16; o > 0; o >>= 1) v += __shfl_xor(v, o, 32);
  return v;
}

// ---------------- layer-1 GEMMs: xl = x@W1l + b1l, xr = x@W1r + b1r ----------
// Block = 80 output rows (5 M-tiles) staged in LDS; 8 waves: waves 0-3 -> xl
// cols [16w,16w+16), waves 4-7 -> xr. Each wave: per K-step load ONE B frag
// and issue 5 independent v_wmma_f32_16x16x4_f32 (one per M-tile) -> 160 wmma
// per wave at a 5:2 wmma:vmem ratio (B reused across the 5 tiles).
__global__ __launch_bounds__(256)
void gat_gemm1(const float* __restrict__ x,
               const float* __restrict__ Wl, const float* __restrict__ bl,
               const float* __restrict__ Wr, const float* __restrict__ br,
               float* __restrict__ xl, float* __restrict__ xr) {
  __shared__ float sx[MTILES * 16 * 128];              // 40 KB A tile (80x128)
  const int m0 = blockIdx.x * (MTILES * 16);
  const float4* xg = (const float4*)(x + (size_t)m0 * 128);
  float4* sx4 = (float4*)sx;
  for (int i = threadIdx.x; i < MTILES * 16 * 32; i += blockDim.x) sx4[i] = xg[i];
  __syncthreads();

  const int wave = threadIdx.x >> 5;
  const int lane = threadIdx.x & 31;
  const bool left = wave < 4;
  const int n0 = (wave & 3) * 16;
  const float* W    = left ? Wl : Wr;
  const float* bias = left ? bl : br;
  float*       out  = left ? xl : xr;

  const int mr = lane & 15;          // M for A frags / N for B,D frags
  const int kh = (lane >> 4) * 2;    // K-pair select (0 or 2)

  v8f acc[MTILES];
#pragma unroll
  for (int t = 0; t < MTILES; ++t) acc[t] = (v8f){};

#pragma unroll
  for (int k = 0; k < 128; k += 4) {
    v2f b;
    b.x = W[(size_t)(k + kh) * HIDF + n0 + mr];
    b.y = W[(size_t)(k + kh + 1) * HIDF + n0 + mr];
#pragma unroll
    for (int t = 0; t < MTILES; ++t) {
      v2f a;
      a.x = sx[(t * 16 + mr) * 128 + k + kh];
      a.y = sx[(t * 16 + mr) * 128 + k + kh + 1];
      acc[t] = __builtin_amdgcn_wmma_f32_16x16x4_f32(false, a, false, b,
                                                     (short)0, acc[t], false, false);
    }
  }
  const float bv = bias[n0 + mr];
  const int half = lane >> 4;
#pragma unroll
  for (int t = 0; t < MTILES; ++t) {
#pragma unroll
    for (int r = 0; r < 8; ++r) {
      int row = m0 + t * 16 + r + 8 * half;  // D: VGPR r -> M=r / M=8+r
      out[(size_t)row * HIDF + n0 + mr] = acc[t][r] + bv;
    }
  }
}

// -------- layer-1 edge score: e = att . lrelu(xl[src]+xr[dst]), seg-max -----
__global__ __launch_bounds__(256)
void gat_edge_score1(const int* __restrict__ ei,
                     const float* __restrict__ xl, const float* __restrict__ xr,
                     const float* __restrict__ att,
                     float* __restrict__ ebuf, unsigned* __restrict__ mmax) {
  const int e = blockIdx.x * 8 + (threadIdx.x >> 5);   // one wave32 per edge
  if (e >= ET) return;
  const int lane = threadIdx.x & 31;
  int src, dst;
  if (e < EE) { src = ei[e]; dst = ei[EE + e]; } else { src = dst = e - EE; }
  const float4 a = *(const float4*)(xl + (size_t)src * 128 + lane * 4);
  const float4 b = *(const float4*)(xr + (size_t)dst * 128 + lane * 4);
  const float4 t = *(const float4*)(att + lane * 4);
  float s = t.x * lrelu(a.x + b.x) + t.y * lrelu(a.y + b.y)
          + t.z * lrelu(a.z + b.z) + t.w * lrelu(a.w + b.w);
  s = wred(s);
  if (lane == 0) {
    ebuf[e] = s;
    atomicMax(mmax + dst, fenc(s));
  }
}

// -------- shared: ex = exp(e - max[dst]); denom[dst] += ex ------------------
__global__ void gat_edge_exp(const int* __restrict__ ei, float* __restrict__ ebuf,
                             const unsigned* __restrict__ mmax,
                             float* __restrict__ denom) {
  const int e = blockIdx.x * blockDim.x + threadIdx.x;
  if (e >= ET) return;
  const int dst = (e < EE) ? ei[EE + e] : (e - EE);
  float ex = expf(ebuf[e] - fdec(mmax[dst]));
  ebuf[e] = ex;
  atomicAdd(denom + dst, ex);
}

// -------- layer-1 aggregate: agg[dst] += alpha * xl[src] --------------------
__global__ __launch_bounds__(256)
void gat_edge_agg1(const int* __restrict__ ei, const float* __restrict__ ebuf,
                   const float* __restrict__ denom, const float* __restrict__ xl,
                   float* __restrict__ agg) {
  const int e = blockIdx.x * 8 + (threadIdx.x >> 5);
  if (e >= ET) return;
  const int lane = threadIdx.x & 31;
  int src, dst;
  if (e < EE) { src = ei[e]; dst = ei[EE + e]; } else { src = dst = e - EE; }
  const float alpha = ebuf[e] / denom[dst];
  const float4 a = *(const float4*)(xl + (size_t)src * 128 + lane * 4);
  float* o = agg + (size_t)dst * 128 + lane * 4;
  atomicAdd(o + 0, alpha * a.x);
  atomicAdd(o + 1, alpha * a.y);
  atomicAdd(o + 2, alpha * a.z);
  atomicAdd(o + 3, alpha * a.w);
}

__global__ void gat_relu_bias(float* __restrict__ h, const float* __restrict__ bias) {
  const int i = blockIdx.x * blockDim.x + threadIdx.x;
  if (i >= NN * 128) return;
  float v = h[i] + bias[i & 127];
  h[i] = v > 0.f ? v : 0.f;
}

// -------- layer-2 matvecs (OUT=1): hl = h@W2l + b2l, hr = h@W2r + b2r -------
__global__ __launch_bounds__(256)
void gat_matvec2(const float* __restrict__ h,
                 const float* __restrict__ W2l, const float* __restrict__ b2l,
                 const float* __restrict__ W2r, const float* __restrict__ b2r,
                 float* __restrict__ hl, float* __restrict__ hr) {
  const int n = blockIdx.x * 8 + (threadIdx.x >> 5);
  if (n >= NN) return;
  const int lane = threadIdx.x & 31;
  const float4 hv = *(const float4*)(h + (size_t)n * 128 + lane * 4);
  const float4 wl = *(const float4*)(W2l + lane * 4);
  const float4 wr = *(const float4*)(W2r + lane * 4);
  float sl = hv.x * wl.x + hv.y * wl.y + hv.z * wl.z + hv.w * wl.w;
  float sr = hv.x * wr.x + hv.y * wr.y + hv.z * wr.z + hv.w * wr.w;
  sl = wred(sl);
  sr = wred(sr);
  if (lane == 0) { hl[n] = sl + b2l[0]; hr[n] = sr + b2r[0]; }
}

__global__ void gat_edge_score2(const int* __restrict__ ei,
                                const float* __restrict__ hl, const float* __restrict__ hr,
                                const float* __restrict__ att2,
                                float* __restrict__ ebuf, unsigned* __restrict__ mmax) {
  const int e = blockIdx.x * blockDim.x + threadIdx.x;
  if (e >= ET) return;
  int src, dst;
  if (e < EE) { src = ei[e]; dst = ei[EE + e]; } else { src = dst = e - EE; }
  float s = att2[0] * lrelu(hl[src] + hr[dst]);
  ebuf[e] = s;
  atomicMax(mmax + dst, fenc(s));
}

__global__ void gat_edge_agg2(const int* __restrict__ ei, const float* __restrict__ ebuf,
                              const float* __restrict__ denom,
                              const float* __restrict__ hl, float* __restrict__ agg2) {
  const int e = blockIdx.x * blockDim.x + threadIdx.x;
  if (e >= ET) return;
  int src, dst;
  if (e < EE) { src = ei[e]; dst = ei[EE + e]; } else { src = dst = e - EE; }
  atomicAdd(agg2 + dst, (ebuf[e] / denom[dst]) * hl[src]);
}

__global__ void gat_final(const float* __restrict__ agg2, const float* __restrict__ bias2,
                          float* __restrict__ out) {
  const int n = blockIdx.x * blockDim.x + threadIdx.x;
  if (n >= NN) return;
  float v = 1.f / (1.f + expf(-(agg2[n] + bias2[0])));
  out[n] = v;
  out[NN + n] = v;
  out[2 * NN + n] = v;
}

extern "C" void kernel_launch(void* const* d_in, const int* in_sizes, int n_in,
                              void* d_out, int out_size, void* d_ws, size_t ws_size,
                              hipStream_t stream) {
  const float* x     = (const float*)d_in[0];
  const int*   ei    = (const int*)d_in[1];
  const float* W1l   = (const float*)d_in[2];
  const float* b1l   = (const float*)d_in[3];
  const float* W1r   = (const float*)d_in[4];
  const float* b1r   = (const float*)d_in[5];
  const float* att1  = (const float*)d_in[6];
  const float* bias1 = (const float*)d_in[7];
  const float* W2l   = (const float*)d_in[8];
  const float* b2l   = (const float*)d_in[9];
  const float* W2r   = (const float*)d_in[10];
  const float* b2r   = (const float*)d_in[11];
  const float* att2  = (const float*)d_in[12];
  const float* bias2 = (const float*)d_in[13];
  float* out = (float*)d_out;

  float* ws = (float*)d_ws;
  float*    xl    = ws;                                  // N*128
  float*    xr    = xl + (size_t)NN * 128;               // N*128
  float*    agg   = xr + (size_t)NN * 128;               // N*128 (h in-place)
  float*    ebuf  = agg + (size_t)NN * 128;              // ET
  unsigned* mmax  = (unsigned*)(ebuf + ET);              // N
  float*    denom = (float*)mmax + NN;                   // N
  float*    hl    = denom + NN;                          // N
  float*    hr    = hl + NN;                             // N
  float*    agg2  = hr + NN;                             // N

  // ---- layer 1 ----
  hipMemsetAsync(agg,  0, sizeof(float) * (size_t)NN * 128, stream);
  hipMemsetAsync(mmax, 0, sizeof(unsigned) * NN, stream);  // 0 == fenc(below all finites)
  hipMemsetAsync(denom, 0, sizeof(float) * NN, stream);

  gat_gemm1<<<NN / (MTILES * 16), 256, 0, stream>>>(x, W1l, b1l, W1r, b1r, xl, xr);
  gat_edge_score1<<<(ET + 7) / 8, 256, 0, stream>>>(ei, xl, xr, att1, ebuf, mmax);
  gat_edge_exp<<<(ET + 255) / 256, 256, 0, stream>>>(ei, ebuf, mmax, denom);
  gat_edge_agg1<<<(ET + 7) / 8, 256, 0, stream>>>(ei, ebuf, denom, xl, agg);
  gat_relu_bias<<<(NN * 128 + 255) / 256, 256, 0, stream>>>(agg, bias1);

  // ---- layer 2 ----
  gat_matvec2<<<(NN + 7) / 8, 256, 0, stream>>>(agg, W2l, b2l, W2r, b2r, hl, hr);

  hipMemsetAsync(mmax, 0, sizeof(unsigned) * NN, stream);
  hipMemsetAsync(denom, 0, sizeof(float) * NN, stream);
  hipMemsetAsync(agg2, 0, sizeof(float) * NN, stream);

  gat_edge_score2<<<(ET + 255) / 256, 256, 0, stream>>>(ei, hl, hr, att2, ebuf, mmax);
  gat_edge_exp<<<(ET + 255) / 256, 256, 0, stream>>>(ei, ebuf, mmax, denom);
  gat_edge_agg2<<<(ET + 255) / 256, 256, 0, stream>>>(ei, ebuf, denom, hl, agg2);
  gat_final<<<(NN + 255) / 256, 256, 0, stream>>>(agg2, bias2, out);
}